// SimpleDetectionLoss_31112743092297
// MI455X (gfx1250) — compile-verified
//
#include <hip/hip_runtime.h>
#include <hip/hip_bf16.h>
#include <math.h>

// Problem constants (from reference)
constexpr int   Bn = 8;
constexpr int   Nn = 262144;
constexpr int   Mn = 100;
constexpr int   Cn = 7;
constexpr float EPSF = 1e-7f;

// Tiling
constexpr int NBLK = 64;            // blocks per image (kernel A)
constexpr int APB  = Nn / NBLK;     // 4096 anchors per block
constexpr int NW   = 8;             // waves per 256-thread block
constexpr int CPW  = APB / 16 / NW; // 32 chunks of 16 anchors per wave
constexpr int MT   = 7;             // 7 m-tiles of 16 -> 112 (>= Mn)

typedef __attribute__((ext_vector_type(2))) float v2f;
typedef __attribute__((ext_vector_type(8))) float v8f;

// ---------------------------------------------------------------------------
// Kernel A: one WMMA produces d^2 directly via K=4 augmentation:
//   A row(n)  = ( ax,   ay,  |a|^2, 1    )      (lanes 0-15 K0,K1; lanes 16-31 K2,K3)
//   B col(m)  = (-2gx, -2gy, 1,     |gc|^2)
//   D[n][m]   = |gc|^2 + |a|^2 - 2*(gc . a) = d^2
// ---------------------------------------------------------------------------
__global__ __launch_bounds__(256) void kernelA(
    const float* __restrict__ pred_cls,
    const float* __restrict__ anchors,
    const float* __restrict__ gt_boxes,
    unsigned long long* __restrict__ ws_arg,   // [Bn][NBLK][Mn] packed (d2bits<<32 | n)
    float* __restrict__ ws_neg,                // [Bn][NBLK]
    float* __restrict__ ws_soft)               // [Bn][NBLK]
{
  __shared__ unsigned long long lds_arg[NW * Mn];
  __shared__ float lds_red[256];
  __shared__ int   lds_cnt[NW];

  const int  b    = blockIdx.y;
  const int  blk  = blockIdx.x;
  const int  tid  = threadIdx.x;
  const int  lane = tid & 31;
  const int  wave = tid >> 5;
  const int  col  = lane & 15;
  const bool hif  = lane >= 16;

  // Preload GT-center tiles as augmented B operands.
  // Pads (m>=100): gx=gy=0, |gc|^2=1e30 -> huge d2, never wins min-over-m.
  float tb0[MT], tb1[MT];
#pragma unroll
  for (int t = 0; t < MT; ++t) {
    const int m = t * 16 + col;
    float gx = 0.f, gy = 0.f, g2 = 1e30f;
    if (m < Mn) {
      const float* g = gt_boxes + ((size_t)b * Mn + m) * 4;
      gx = (g[0] + g[2]) * 0.5f;
      gy = (g[1] + g[3]) * 0.5f;
      g2 = gx * gx + gy * gy;
    }
    tb0[t] = hif ? 1.0f : (-2.0f * gx);   // K2 = 1      | K0 = -2*gcx
    tb1[t] = hif ? g2   : (-2.0f * gy);   // K3 = |gc|^2 | K1 = -2*gcy
  }

  float    bestk[MT];
  unsigned bestn[MT];
#pragma unroll
  for (int t = 0; t < MT; ++t) { bestk[t] = 3.4e38f; bestn[t] = 0u; }

  int       negcnt = 0;               // # anchors with min_m d2 > 9 (wave-uniform)
  const int nhalf  = hif ? 8 : 0;
  const int base0  = blk * APB + wave * (CPW * 16);

  // software-pipelined anchor load (lane 16+c loads same anchor as lane c)
  float2 apr = *reinterpret_cast<const float2*>(anchors + (size_t)(base0 + col) * 2);

  for (int ci = 0; ci < CPW; ++ci) {
    const int nb = base0 + ci * 16;
    const float ax = apr.x, ay = apr.y;
    if (ci + 1 < CPW)
      apr = *reinterpret_cast<const float2*>(anchors + (size_t)(nb + 16 + col) * 2);

    const float aa = fmaf(ax, ax, ay * ay);
    v2f A;
    A.x = hif ? aa   : ax;   // K2 = |a|^2 | K0 = ax
    A.y = hif ? 1.0f : ay;   // K3 = 1     | K1 = ay

    unsigned nr[8];
#pragma unroll
    for (int r = 0; r < 8; ++r) nr[r] = (unsigned)(nb + nhalf + r);

    float negmin[8];
#pragma unroll
    for (int r = 0; r < 8; ++r) negmin[r] = 3.4e38f;

#pragma unroll
    for (int t = 0; t < MT; ++t) {
      v2f Bv; Bv.x = tb0[t]; Bv.y = tb1[t];
      v8f d2 = {0.f, 0.f, 0.f, 0.f, 0.f, 0.f, 0.f, 0.f};
      d2 = __builtin_amdgcn_wmma_f32_16x16x4_f32(false, A, false, Bv,
                                                 (short)0, d2, false, false);
#pragma unroll
      for (int r = 0; r < 8; ++r) {
        negmin[r] = fminf(negmin[r], d2[r]);
        if (d2[r] < bestk[t]) { bestk[t] = d2[r]; bestn[t] = nr[r]; } // strict <: first idx
      }
    }
    // per-anchor "all m beyond 3.0" predicate via ballot + scalar mask tests
    // (min_m clip(d2,0) > 9  <=>  min_m d2 > 9, since 9 > 0)
#pragma unroll
    for (int r = 0; r < 8; ++r) {
      const unsigned mask = (unsigned)__ballot(negmin[r] > 9.0f);
      negcnt += ((mask & 0xffffu) == 0xffffu) ? 1 : 0;   // anchor nb + r
      negcnt += ((mask >> 16)     == 0xffffu) ? 1 : 0;   // anchor nb + 8 + r
    }
  }

  // combine the two halves (same m, anchors differ by +8)
#pragma unroll
  for (int t = 0; t < MT; ++t) {
    const float    ok = __shfl_xor(bestk[t], 16, 32);
    const unsigned on = __shfl_xor(bestn[t], 16, 32);
    if (ok < bestk[t] || (ok == bestk[t] && on < bestn[t])) { bestk[t] = ok; bestn[t] = on; }
  }
  if (!hif) {
#pragma unroll
    for (int t = 0; t < MT; ++t) {
      const int m = t * 16 + col;
      if (m < Mn) {
        const float kc = fmaxf(bestk[t], 0.f);  // reference clips before sqrt/argmin
        lds_arg[wave * Mn + m] =
            ((unsigned long long)__float_as_uint(kc) << 32) | (unsigned long long)bestn[t];
      }
    }
  }

  // softplus partial over this block's pred_cls slice (fixed order per thread)
  float ss = 0.f;
  const float* cp = pred_cls + ((size_t)b * Nn + (size_t)blk * APB) * Cn;
  for (int i = tid; i < APB * Cn; i += 256) {
    const float x = cp[i];
    ss += fmaxf(x, 0.f) + log1pf(expf(-fabsf(x)));   // logaddexp(0, x)
  }
  lds_red[tid] = ss;
  if (lane == 0) lds_cnt[wave] = negcnt;
  __syncthreads();
  for (int s = 128; s > 0; s >>= 1) {
    if (tid < s) lds_red[tid] += lds_red[tid + s];
    __syncthreads();
  }
  if (tid == 0) {
    ws_soft[b * NBLK + blk] = lds_red[0];
    int c = 0;
    for (int w = 0; w < NW; ++w) c += lds_cnt[w];
    // sum over block's anchors of (neg ? 1 : 0.1) = 0.1*APB + 0.9*count
    ws_neg[b * NBLK + blk] = 0.1f * (float)APB + 0.9f * (float)c;
  }

  // argmin: reduce the 8 wave partials, emit per-block packed minimum
  if (tid < Mn) {
    unsigned long long best = ~0ull;
    for (int w = 0; w < NW; ++w) {
      const unsigned long long v = lds_arg[w * Mn + tid];
      if (v < best) best = v;   // packed compare: min d2, ties -> smallest n
    }
    ws_arg[((size_t)b * NBLK + blk) * Mn + tid] = best;
  }
}

// ---------------------------------------------------------------------------
// Kernel B: per-image epilogue (one block per image)
// ---------------------------------------------------------------------------
__global__ __launch_bounds__(128) void kernelB(
    const float* __restrict__ pred_cls,
    const float* __restrict__ pred_reg,
    const float* __restrict__ anchors,
    const float* __restrict__ strides,
    const float* __restrict__ gt_boxes,
    const int*   __restrict__ gt_labels,
    const unsigned long long* __restrict__ ws_arg,
    const float* __restrict__ ws_neg,
    const float* __restrict__ ws_soft,
    float* __restrict__ ws_lc,
    float* __restrict__ ws_lb)
{
  __shared__ unsigned assigned[Mn];
  __shared__ unsigned sorted_[Mn];
  __shared__ float gx_[Mn], gy_[Mn], gg_[Mn];
  __shared__ int   cid_[Mn];
  __shared__ float red[128];
  __shared__ int   redi[128];
  __shared__ float s_base[2];   // {negsum, softsum}

  const int b = blockIdx.x;
  const int t = threadIdx.x;

  if (t < Mn) {
    unsigned long long best = ~0ull;
    for (int blk = 0; blk < NBLK; ++blk) {
      const unsigned long long v = ws_arg[((size_t)b * NBLK + blk) * Mn + t];
      if (v < best) best = v;
    }
    assigned[t] = (unsigned)(best & 0xffffffffu);
    const float* g = gt_boxes + ((size_t)b * Mn + t) * 4;
    const float gx = (g[0] + g[2]) * 0.5f, gy = (g[1] + g[3]) * 0.5f;
    gx_[t] = gx; gy_[t] = gy; gg_[t] = gx * gx + gy * gy;
    int c = gt_labels[b * Mn + t] - 1;
    cid_[t] = min(max(c, 0), Cn - 1);
  }
  if (t == 0) {
    float ns = 0.f, ssum = 0.f;
    for (int blk = 0; blk < NBLK; ++blk) {
      ns   += ws_neg[b * NBLK + blk];
      ssum += ws_soft[b * NBLK + blk];
    }
    s_base[0] = ns; s_base[1] = ssum;
  }
  __syncthreads();

  float clsg = 0.f, poscorr = 0.f; int firsta = 0;
  if (t < Mn) {
    const unsigned a = assigned[t];
    const int      c = cid_[t];
    bool fpair = true, fanch = true;
    for (int j = 0; j < t; ++j) {
      if (assigned[j] == a) { fanch = false; if (cid_[j] == c) fpair = false; }
    }
    if (fpair) clsg = pred_cls[((size_t)b * Nn + a) * Cn + c]; // distinct tgt scatter pos
    if (fanch) {                                              // distinct positive anchor
      firsta = 1;
      const float ax = anchors[(size_t)a * 2], ay = anchors[(size_t)a * 2 + 1];
      const float aa = ax * ax + ay * ay;
      float mind = 3.4e38f;
      for (int m = 0; m < Mn; ++m) {
        const float d2 = gg_[m] + aa - 2.0f * (gx_[m] * ax + gy_[m] * ay);
        mind = fminf(mind, fmaxf(d2, 0.f));
      }
      poscorr = (mind > 9.0f) ? 1.0f : 0.1f;  // sw goes 1x -> 2x on positives
    }
  }
  red[t] = clsg; redi[t] = firsta;
  __syncthreads();
  for (int s = 64; s > 0; s >>= 1) {
    if (t < s) { red[t] += red[t + s]; redi[t] += redi[t + s]; }
    __syncthreads();
  }
  const float clssum = red[0];
  const int   pcnt   = redi[0];
  __syncthreads();
  red[t] = poscorr;
  __syncthreads();
  for (int s = 64; s > 0; s >>= 1) {
    if (t < s) red[t] += red[t + s];
    __syncthreads();
  }
  const float corr = red[0];
  __syncthreads();

  // stable rank sort of assigned (jnp.sort equivalent)
  if (t < Mn) {
    const unsigned a = assigned[t];
    int rank = 0;
    for (int j = 0; j < Mn; ++j) {
      const unsigned aj = assigned[j];
      rank += (aj < a || (aj == a && j < t)) ? 1 : 0;
    }
    sorted_[rank] = a;
  }
  __syncthreads();

  const bool use_sorted = (pcnt == Mn);  // sum(pos_mask) == M  <=>  all distinct
  float gl = 0.f;
  if (t < Mn) {
    const unsigned a = use_sorted ? sorted_[t] : assigned[t];
    const float s  = strides[a];
    const float* rg = pred_reg + ((size_t)b * Nn + a) * 4;
    const float cx = anchors[(size_t)a * 2]     + rg[0] * s;
    const float cy = anchors[(size_t)a * 2 + 1] + rg[1] * s;
    const float w  = expf(rg[2]) * s;
    const float h  = expf(rg[3]) * s;
    const float p0 = cx - w * 0.5f, p1 = cy - h * 0.5f;
    const float p2 = cx + w * 0.5f, p3 = cy + h * 0.5f;
    const float* g = gt_boxes + ((size_t)b * Mn + t) * 4;
    const float g0 = g[0], g1 = g[1], g2v = g[2], g3 = g[3];
    const float ap = (p2 - p0) * (p3 - p1);
    const float ag = (g2v - g0) * (g3 - g1);
    const float ltx = fmaxf(p0, g0), lty = fmaxf(p1, g1);
    const float rbx = fminf(p2, g2v), rby = fminf(p3, g3);
    const float wi = fmaxf(rbx - ltx, 0.f), hi2 = fmaxf(rby - lty, 0.f);
    const float inter = wi * hi2;
    const float uni   = ap + ag - inter;
    const float iou   = inter / (uni + EPSF);
    const float lcx = fminf(p0, g0), lcy = fminf(p1, g1);
    const float rcx = fmaxf(p2, g2v), rcy = fmaxf(p3, g3);
    const float wc = fmaxf(rcx - lcx, 0.f), hc = fmaxf(rcy - lcy, 0.f);
    const float ac = wc * hc;
    const float giou = iou - (ac - uni) / (ac + EPSF);
    gl = 1.0f - giou;
  }
  red[t] = gl;
  __syncthreads();
  for (int s = 64; s > 0; s >>= 1) {
    if (t < s) red[t] += red[t + s];
    __syncthreads();
  }
  if (t == 0) {
    const float mean_sw = (s_base[0] + corr) / (float)Nn;
    const float bce     = (s_base[1] - clssum) / ((float)Nn * (float)Cn);
    ws_lc[b] = bce * mean_sw;
    ws_lb[b] = red[0] / (float)Mn;
  }
}

// ---------------------------------------------------------------------------
// Kernel C: final scalar
// ---------------------------------------------------------------------------
__global__ void kernelC(const float* __restrict__ ws_lc,
                        const float* __restrict__ ws_lb,
                        float* __restrict__ out)
{
  if (threadIdx.x == 0 && blockIdx.x == 0) {
    float lc = 0.f, lb = 0.f;
    for (int b = 0; b < Bn; ++b) { lc += ws_lc[b]; lb += ws_lb[b]; }
    out[0] = lc / (float)Bn + 2.0f * (lb / (float)Bn);
  }
}

// ---------------------------------------------------------------------------
extern "C" void kernel_launch(void* const* d_in, const int* in_sizes, int n_in,
                              void* d_out, int out_size, void* d_ws, size_t ws_size,
                              hipStream_t stream) {
  (void)in_sizes; (void)n_in; (void)out_size; (void)ws_size;
  const float* pred_cls  = (const float*)d_in[0];
  const float* pred_reg  = (const float*)d_in[1];
  const float* anchors   = (const float*)d_in[2];
  const float* strides   = (const float*)d_in[3];
  const float* gt_boxes  = (const float*)d_in[4];
  const int*   gt_labels = (const int*)d_in[5];
  float* out = (float*)d_out;

  // workspace carve (all slots fully rewritten every call; no stale reads)
  unsigned long long* ws_arg = (unsigned long long*)d_ws;          // Bn*NBLK*Mn u64
  float* ws_neg  = (float*)(ws_arg + (size_t)Bn * NBLK * Mn);      // Bn*NBLK
  float* ws_soft = ws_neg + Bn * NBLK;                              // Bn*NBLK
  float* ws_lc   = ws_soft + Bn * NBLK;                             // Bn
  float* ws_lb   = ws_lc + Bn;                                      // Bn

  kernelA<<<dim3(NBLK, Bn), 256, 0, stream>>>(pred_cls, anchors, gt_boxes,
                                              ws_arg, ws_neg, ws_soft);
  kernelB<<<dim3(Bn), 128, 0, stream>>>(pred_cls, pred_reg, anchors, strides,
                                        gt_boxes, gt_labels,
                                        ws_arg, ws_neg, ws_soft, ws_lc, ws_lb);
  kernelC<<<1, 32, 0, stream>>>(ws_lc, ws_lb, out);
}